// TorchSMoE_SMoE_38225208934844
// MI455X (gfx1250) — compile-verified
//
#include <hip/hip_runtime.h>

typedef __attribute__((ext_vector_type(2))) float v2f;
typedef __attribute__((ext_vector_type(8))) float v8f;

// Reference constants: N_KERNELS=4, BLOCK_SIZE=8 -> 64 grid points, 28 params/row.
#define PARAMS_PER_ROW 28
#define INV7 0.14285714285714285f
// -0.5 * log2(e): folded into the polynomial coefficients so the WMMA result
// feeds v_exp_f32 (base-2) directly, removing all per-element scaling VALU.
#define NEG_HALF_LOG2E (-0.72134752044448169f)

// ---------------------------------------------------------------------------
// Main kernel: one wave32 handles 4 batch rows (16 WMMA rows = 4 batches x 4
// mixture components). quad-form evaluation is a (16 x 64) = (16x8)*(8x64)
// f32 GEMM against the constant monomial basis [1, x, y, x^2, xy, y^2, 0, 0],
// executed as 2 x V_WMMA_F32_16X16X4_F32 per 16-column tile (4 tiles).
// ---------------------------------------------------------------------------
__global__ __launch_bounds__(256) void smoe_wmma_kernel(
    const float* __restrict__ arr, float* __restrict__ out) {
  const int lane  = threadIdx.x & 31;
  const int wave  = (blockIdx.x << 3) + (threadIdx.x >> 5); // 8 waves/block
  const int bbase = wave << 2;                              // 4 batches/wave

  // ---- per-lane coefficient generation for WMMA row m = lane & 15 ----
  const int  m  = lane & 15;       // tile row: 4*localBatch + k
  const int  lb = m >> 2;          // local batch 0..3
  const int  k  = m & 3;           // mixture component 0..3
  const bool hi = lane >= 16;      // upper half-wave holds K=2,3 (and rows 8..15 of C)

  const float* p = arr + (size_t)(bbase + lb) * PARAMS_PER_ROW;
  const float cx  = p[k];
  const float cy  = p[4 + k];
  const float nue = p[8 + k];
  // tril of 2x2: a00 = v0, a10 = v2, a11 = v3 (v1 dropped)
  const float a00 = p[12 + 4 * k + 0];
  const float a10 = p[12 + 4 * k + 2];
  const float a11 = p[12 + 4 * k + 3];

  // quad(x,y) = (a00*x + a10*y - u)^2 + (a11*y - t)^2
  //           = c0 + c1*x + c2*y + c3*x^2 + c4*xy + c5*y^2
  // We directly compute s*c_i with s = -0.5*log2(e), so the GEMM produces
  // log2(weight) and v_exp_f32 applies with no further scaling.
  const float u  = a00 * cx + a10 * cy;
  const float t  = a11 * cy;
  const float s  = NEG_HALF_LOG2E;
  const float c0 = s * (u * u + t * t);
  const float c1 = s * (-2.0f * a00 * u);
  const float c2 = s * (-2.0f * a10 * u - 2.0f * a11 * t);
  const float c3 = s * (a00 * a00);
  const float c4 = s * (2.0f * a00 * a10);
  const float c5 = s * (a10 * a10 + a11 * a11);

  // A operands (16x4 f32 layout: lanes 0-15 -> {K0,K1}, lanes 16-31 -> {K2,K3})
  v2f A0;  A0.x = hi ? c2   : c0;  A0.y = hi ? c3   : c1;   // K = 0..3
  v2f A1;  A1.x = hi ? 0.0f : c4;  A1.y = hi ? 0.0f : c5;   // K = 4..7 (6,7 pad)

  // Broadcast nue for the 8 C-rows this lane will hold (rows rbase..rbase+7).
  const int rbase = hi ? 8 : 0;
  float nue_r[8];
#pragma unroll
  for (int v = 0; v < 8; ++v) nue_r[v] = __shfl(nue, rbase + v, 32);

  const int lb_lo = hi ? 2 : 0;  // C rows 0-3/8-11 -> batches 0/2
  const int n16   = lane & 15;

#pragma unroll
  for (int tile = 0; tile < 4; ++tile) {
    const int   col = (tile << 4) + n16;          // grid point 0..63
    const float px  = (float)(col >> 3) * INV7;   // x = (c/8)/7
    const float py  = (float)(col & 7) * INV7;    // y = (c%8)/7

    // B operands (4x16 f32 layout mirrors A): rows are basis functions.
    v2f B0, B1;
    if (!hi) { B0.x = 1.0f;     B0.y = px;       B1.x = px * py; B1.y = py * py; }
    else     { B0.x = py;       B0.y = px * px;  B1.x = 0.0f;    B1.y = 0.0f;    }

    v8f acc = {};
    acc = __builtin_amdgcn_wmma_f32_16x16x4_f32(false, A0, false, B0,
                                                (short)0, acc, false, false);
    acc = __builtin_amdgcn_wmma_f32_16x16x4_f32(false, A1, false, B1,
                                                (short)0, acc, false, false);

    // acc[v] = log2(weight) for row (rbase+v) at this column.
    float e[8];
#pragma unroll
    for (int v = 0; v < 8; ++v) e[v] = __builtin_amdgcn_exp2f(acc[v]);

    const float den0 = e[0] + e[1] + e[2] + e[3];
    const float den1 = e[4] + e[5] + e[6] + e[7];
    const float num0 = nue_r[0] * e[0] + nue_r[1] * e[1] +
                       nue_r[2] * e[2] + nue_r[3] * e[3];
    const float num1 = nue_r[4] * e[4] + nue_r[5] * e[5] +
                       nue_r[6] * e[6] + nue_r[7] * e[7];

    float r0 = num0 / fmaxf(den0, 1e-8f);
    float r1 = num1 / fmaxf(den1, 1e-8f);
    r0 = fminf(fmaxf(r0, 0.0f), 1.0f);
    r1 = fminf(fmaxf(r1, 0.0f), 1.0f);

    out[(size_t)(bbase + lb_lo) * 64 + col]     = r0;
    out[(size_t)(bbase + lb_lo + 1) * 64 + col] = r1;
  }
}

// ---------------------------------------------------------------------------
// Scalar tail kernel: handles batch rows not covered by full 32-batch blocks
// (unused for BATCH=524288, kept for robustness since WMMA needs full EXEC).
// ---------------------------------------------------------------------------
__global__ void smoe_tail_kernel(const float* __restrict__ arr,
                                 float* __restrict__ out,
                                 int start, int end) {
  const int b = start + blockIdx.x * blockDim.x + threadIdx.x;
  if (b >= end) return;
  const float* p = arr + (size_t)b * PARAMS_PER_ROW;
  float cx[4], cy[4], nue[4], a00[4], a10[4], a11[4];
#pragma unroll
  for (int kk = 0; kk < 4; ++kk) {
    cx[kk]  = p[kk];
    cy[kk]  = p[4 + kk];
    nue[kk] = p[8 + kk];
    a00[kk] = p[12 + 4 * kk + 0];
    a10[kk] = p[12 + 4 * kk + 2];
    a11[kk] = p[12 + 4 * kk + 3];
  }
  for (int c = 0; c < 64; ++c) {
    const float px = (float)(c >> 3) * INV7;
    const float py = (float)(c & 7) * INV7;
    float den = 0.0f, num = 0.0f;
#pragma unroll
    for (int kk = 0; kk < 4; ++kk) {
      const float dx = px - cx[kk];
      const float dy = py - cy[kk];
      const float q0 = dx * a00[kk] + dy * a10[kk];
      const float q1 = dy * a11[kk];
      const float e  = __builtin_amdgcn_exp2f(NEG_HALF_LOG2E * (q0 * q0 + q1 * q1));
      den += e;
      num += nue[kk] * e;
    }
    float r = num / fmaxf(den, 1e-8f);
    out[(size_t)b * 64 + c] = fminf(fmaxf(r, 0.0f), 1.0f);
  }
}

extern "C" void kernel_launch(void* const* d_in, const int* in_sizes, int n_in,
                              void* d_out, int out_size, void* d_ws, size_t ws_size,
                              hipStream_t stream) {
  const float* arr = (const float*)d_in[0];
  float* out = (float*)d_out;
  const int batch = in_sizes[0] / PARAMS_PER_ROW;   // 524288

  const int mainB = batch & ~31;      // 32 batches per 256-thread block
  if (mainB > 0) {
    smoe_wmma_kernel<<<mainB / 32, 256, 0, stream>>>(arr, out);
  }
  if (batch > mainB) {
    const int rem = batch - mainB;
    smoe_tail_kernel<<<(rem + 127) / 128, 128, 0, stream>>>(arr, out, mainB, batch);
  }
}